// Connectome_58703613002216
// MI455X (gfx1250) — compile-verified
//
#include <hip/hip_runtime.h>
#include <hip/hip_bf16.h>
#include <math.h>

// ---------------------------------------------------------------------------
// Problem constants (from the reference): N=8, T=512, V=128*128=16384, R=400
// latent GEMM:  (N*T=4096 x 16384) * (16384 x 400)  -> latent (4096 x 400)
// conn GEMM :   per n: x^T x,  x = (512 x 400) centered+normalized over t
// Outputs flat: cls_token (8*79800) then latent (8*512*400)
// Precision: fp32 throughout via V_WMMA_F32_16X16X4_F32 (matches reference
// numerics; normalization amplifies any A/B truncation error).
// ---------------------------------------------------------------------------

typedef __attribute__((ext_vector_type(2))) float v2f;
typedef __attribute__((ext_vector_type(8))) float v8f;

#define KTOT 16384
#define RTOT 400
#define MTOT 4096     // N*T
#define TDIM 512
#define NBATCH 8
#define TRIU  79800   // 400*399/2
#define LATSZ (TDIM * RTOT) // per-n latent elems = 204800

// ============================ Kernel 1: latent GEMM =========================
// Block tile 128(M) x 80(N), K-step 32. 8 waves; wave w owns rows
// [w*16, w*16+16) and all five 16-col tiles (5 x v8f accumulators).
// LDS rows padded to 33 floats -> conflict-free (33 coprime with 64 banks).
// Ping-pong double buffering: next K-tile's global loads are issued into
// registers before computing on the current LDS buffer, so s_wait_loadcnt
// lands behind 40 WMMAs of work; one barrier per K-step.
#define MB 128
#define NB 80
#define KB 32
#define LDP 33
#define AREG ((MB * KB) / 256)   // 16 staged A floats per thread
#define BREG ((NB * KB) / 256)   // 10 staged B floats per thread

__global__ __launch_bounds__(256) void latent_gemm_kernel(
    const float* __restrict__ A,     // imgs as (4096, 16384)
    const float* __restrict__ Bp,    // parc_weight (400, 16384)
    float* __restrict__ latent)      // (4096, 400)
{
    __shared__ float As[2][MB * LDP];   // As[buf][row*LDP + k]
    __shared__ float Bs[2][NB * LDP];   // Bs[buf][nn*LDP + k]

    const int tid  = threadIdx.x;
    const int wave = tid >> 5;
    const int lane = tid & 31;
    const int g    = lane >> 4;      // half-wave select
    const int c    = lane & 15;      // position within half
    const int m0   = blockIdx.x * MB;
    const int n0   = blockIdx.y * NB;

    v8f acc[5];
#pragma unroll
    for (int i = 0; i < 5; ++i) acc[i] = (v8f){0.f,0.f,0.f,0.f,0.f,0.f,0.f,0.f};

    float ra[AREG], rb[BREG];

    // ---- prologue: stage tile k0=0 ----
#pragma unroll
    for (int i = 0; i < AREG; ++i) {
        int f = tid + i * 256;
        ra[i] = A[(size_t)(m0 + (f >> 5)) * KTOT + (f & 31)];
    }
#pragma unroll
    for (int i = 0; i < BREG; ++i) {
        int f = tid + i * 256;
        rb[i] = Bp[(size_t)(n0 + (f >> 5)) * KTOT + (f & 31)];
    }
#pragma unroll
    for (int i = 0; i < AREG; ++i) {
        int f = tid + i * 256;
        As[0][(f >> 5) * LDP + (f & 31)] = ra[i];
    }
#pragma unroll
    for (int i = 0; i < BREG; ++i) {
        int f = tid + i * 256;
        Bs[0][(f >> 5) * LDP + (f & 31)] = rb[i];
    }

    int cur = 0;
    for (int k0 = 0; k0 < KTOT; k0 += KB) {
        __syncthreads();                       // LDS buffer cur is visible to all waves

        const bool more = (k0 + KB) < KTOT;
        if (more) {
            // Issue next tile's global loads now; data consumed only after
            // the compute below -> loads overlap the 40 WMMAs.
#pragma unroll
            for (int i = 0; i < AREG; ++i) {
                int f = tid + i * 256;
                ra[i] = A[(size_t)(m0 + (f >> 5)) * KTOT + (k0 + KB) + (f & 31)];
            }
#pragma unroll
            for (int i = 0; i < BREG; ++i) {
                int f = tid + i * 256;
                rb[i] = Bp[(size_t)(n0 + (f >> 5)) * KTOT + (k0 + KB) + (f & 31)];
            }
        }

        // ---- compute on buffer `cur` ----
        const float* arow = &As[cur][(wave * 16 + c) * LDP];
        const float* bbase = &Bs[cur][0];
#pragma unroll
        for (int kk = 0; kk < KB; kk += 4) {
            // A frag: M = c, K = kk + 2g + j   (j=0,1 contiguous -> b64 load)
            v2f a = *(const v2f*)(arow + kk + 2 * g);
#pragma unroll
            for (int nt = 0; nt < 5; ++nt) {
                // B frag: N = c, K = kk + 2g + j
                v2f b = *(const v2f*)(bbase + (nt * 16 + c) * LDP + kk + 2 * g);
                acc[nt] = __builtin_amdgcn_wmma_f32_16x16x4_f32(
                    false, a, false, b, (short)0, acc[nt], false, false);
            }
        }

        if (more) {
            // Fill the other buffer; nobody reads it until next barrier.
            const int nxt = cur ^ 1;
#pragma unroll
            for (int i = 0; i < AREG; ++i) {
                int f = tid + i * 256;
                As[nxt][(f >> 5) * LDP + (f & 31)] = ra[i];
            }
#pragma unroll
            for (int i = 0; i < BREG; ++i) {
                int f = tid + i * 256;
                Bs[nxt][(f >> 5) * LDP + (f & 31)] = rb[i];
            }
        }
        cur ^= 1;
    }

    // ---- store: C/D layout  M = v + 8g, N = c ----
#pragma unroll
    for (int nt = 0; nt < 5; ++nt) {
#pragma unroll
        for (int v = 0; v < 8; ++v) {
            int row = m0 + wave * 16 + v + 8 * g;
            int col = n0 + nt * 16 + c;
            latent[(size_t)row * RTOT + col] = acc[nt][v];
        }
    }
}

// ======================= Kernel 2: per-(n,r) stats ==========================
// One thread per column; consecutive lanes -> consecutive r -> coalesced.
__global__ void stats_kernel(const float* __restrict__ latent,
                             float* __restrict__ means,
                             float* __restrict__ scales)
{
    int gid = blockIdx.x * blockDim.x + threadIdx.x;
    if (gid >= NBATCH * RTOT) return;
    int n = gid / RTOT;
    int r = gid % RTOT;
    const float* p = latent + (size_t)n * LATSZ + r;
    float s1 = 0.f, s2 = 0.f;
    for (int t = 0; t < TDIM; ++t) {
        float v = p[(size_t)t * RTOT];
        s1 += v;
        s2 += v * v;
    }
    float mu  = s1 * (1.0f / TDIM);
    float var = s2 - (float)TDIM * mu * mu;   // sum of centered squares
    var = var > 0.f ? var : 0.f;
    means[gid]  = mu;
    scales[gid] = 1.0f / (sqrtf(var) + 1e-6f);
}

// ==================== Kernel 3: conn = x^T x, triu(k=1) =====================
// One wave per (n, tile-pair i<=j). R=400 -> 25 tiles, 325 pairs, 2600 waves.
// Fragments loaded straight from latent (fits in 192MB L2); centering/scaling
// fused into the fragment load (lane's r/s fixed across the whole K loop).
__global__ __launch_bounds__(32) void conn_kernel(
    const float* __restrict__ latent,
    const float* __restrict__ means,
    const float* __restrict__ scales,
    float* __restrict__ cls)             // (8, 79800)
{
    const int bid  = blockIdx.x;
    const int n    = bid / 325;
    int       p    = bid % 325;
    int i = 0;
    while (p >= 25 - i) { p -= 25 - i; ++i; }   // uniform, tiny
    const int j = i + p;

    const int lane = threadIdx.x;
    const int g    = lane >> 4;
    const int c    = lane & 15;

    const int r = i * 16 + c;                    // A-frag row (M)
    const int s = j * 16 + c;                    // B-frag col (N)
    const float muA = means[n * RTOT + r];
    const float scA = scales[n * RTOT + r];
    const float muB = means[n * RTOT + s];
    const float scB = scales[n * RTOT + s];

    const float* base = latent + (size_t)n * LATSZ;

    v8f acc = (v8f){0.f,0.f,0.f,0.f,0.f,0.f,0.f,0.f};
    for (int t0 = 0; t0 < TDIM; t0 += 4) {
        const int ta = t0 + 2 * g;               // K = t0 + 2g + j
        v2f a, b;
        a[0] = (base[(size_t)(ta    ) * RTOT + r] - muA) * scA;
        a[1] = (base[(size_t)(ta + 1) * RTOT + r] - muA) * scA;
        b[0] = (base[(size_t)(ta    ) * RTOT + s] - muB) * scB;
        b[1] = (base[(size_t)(ta + 1) * RTOT + s] - muB) * scB;
        acc = __builtin_amdgcn_wmma_f32_16x16x4_f32(
            false, a, false, b, (short)0, acc, false, false);
    }

    // Emit upper-triangle (rr < ss) entries:
    // triu row-major index = rr*(2R - rr - 1)/2 + (ss - rr - 1)
#pragma unroll
    for (int v = 0; v < 8; ++v) {
        int rr = i * 16 + v + 8 * g;             // C/D: M = v + 8g
        int ss = j * 16 + c;                     // C/D: N = c
        if (ss > rr) {
            int idx = rr * (2 * RTOT - rr - 1) / 2 + (ss - rr - 1);
            cls[(size_t)n * TRIU + idx] = acc[v];
        }
    }
}

// ================================ launch ====================================
extern "C" void kernel_launch(void* const* d_in, const int* in_sizes, int n_in,
                              void* d_out, int out_size, void* d_ws, size_t ws_size,
                              hipStream_t stream)
{
    const float* imgs = (const float*)d_in[0];   // (8,1,512,128,128) f32
    const float* parc = (const float*)d_in[1];   // (400,16384) f32

    float* out    = (float*)d_out;
    float* cls    = out;                          // 8*79800
    float* latent = out + (size_t)NBATCH * TRIU;  // 8*512*400

    float* means  = (float*)d_ws;                 // 3200 f32
    float* scales = means + NBATCH * RTOT;        // 3200 f32

    dim3 gA(MTOT / MB, RTOT / NB);                // (32, 5)
    latent_gemm_kernel<<<gA, 256, 0, stream>>>(imgs, parc, latent);

    int nstats = NBATCH * RTOT;
    stats_kernel<<<(nstats + 255) / 256, 256, 0, stream>>>(latent, means, scales);

    conn_kernel<<<NBATCH * 325, 32, 0, stream>>>(latent, means, scales, cls);
}